// Net_44229573214969
// MI455X (gfx1250) — compile-verified
//
#include <hip/hip_runtime.h>
#include <math.h>

typedef float v2f __attribute__((ext_vector_type(2)));
typedef float v8f __attribute__((ext_vector_type(8)));

#define NNODES 100000
#define NEDGES 3200000
#define FIN    128
#define HDIM   16
#define CDIM   2

// ---------------------------------------------------------------------------
// 1) init: deg = 1 (self loop), zero both aggregation buffers
// ---------------------------------------------------------------------------
__global__ void k_init(float* __restrict__ deg, float* __restrict__ agg1,
                       float* __restrict__ agg2) {
    int i = blockIdx.x * blockDim.x + threadIdx.x;
    if (i < NNODES * HDIM) agg1[i] = 0.0f;
    if (i < NNODES * CDIM) agg2[i] = 0.0f;
    if (i < NNODES)        deg[i]  = 1.0f;
}

// ---------------------------------------------------------------------------
// 2) degree at target nodes (edge weight 1)
// ---------------------------------------------------------------------------
__global__ void k_degree(const int* __restrict__ dst, float* __restrict__ deg) {
    int e = blockIdx.x * blockDim.x + threadIdx.x;
    if (e < NEDGES) atomicAdd(&deg[dst[e]], 1.0f);
}

// ---------------------------------------------------------------------------
// 3) deg -> 1/sqrt(deg) in place
// ---------------------------------------------------------------------------
__global__ void k_rsqrt(float* __restrict__ deg) {
    int i = blockIdx.x * blockDim.x + threadIdx.x;
    if (i < NNODES) deg[i] = rsqrtf(deg[i]);
}

// ---------------------------------------------------------------------------
// 4) xw = x @ W1 via V_WMMA_F32_16X16X4_F32 (exact f32 math)
//    One wave per 16-row tile; K swept in chunks of 4 (32 WMMAs/tile).
//    A 16x4 f32 layout (ISA 7.12.2): lanes 0-15 hold K=k0,k0+1; lanes 16-31
//    hold K=k0+2,k0+3 (row M = lane&15 striped across the half-waves).
//    B 4x16 mirrors A with N = lane&15.  C/D: VGPR r -> (M = r + 8*(lane>>4),
//    N = lane&15).
// ---------------------------------------------------------------------------
__global__ void k_gemm1(const float* __restrict__ x, const float* __restrict__ W1,
                        float* __restrict__ xw) {
    const int lane = threadIdx.x & 31;
    const int wave = threadIdx.x >> 5;
    const int tile = blockIdx.x * (blockDim.x >> 5) + wave;
    if (tile >= NNODES / 16) return;          // wave-uniform: EXEC stays all-1s

    const int m  = lane & 15;                 // row within tile (A) / col (B)
    const int hf = lane >> 4;                 // half-wave selects K pair
    const float* xr = x  + (size_t)(tile * 16 + m) * FIN + 2 * hf;
    const float* wc = W1 + (size_t)(2 * hf) * HDIM + m;

    v8f acc = {};
#pragma unroll
    for (int k0 = 0; k0 < FIN; k0 += 4) {
        v2f a, b;
        a.x = xr[k0];
        a.y = xr[k0 + 1];
        b.x = wc[(size_t)k0 * HDIM];
        b.y = wc[(size_t)(k0 + 1) * HDIM];
        // (neg_a, A, neg_b, B, c_mod, C, reuse_a, reuse_b)
        acc = __builtin_amdgcn_wmma_f32_16x16x4_f32(
            false, a, false, b, (short)0, acc, false, false);
    }

    float* o = xw + (size_t)(tile * 16 + 8 * hf) * HDIM + m;
#pragma unroll
    for (int r = 0; r < 8; ++r) o[(size_t)r * HDIM] = acc[r];
}

// ---------------------------------------------------------------------------
// 5) layer-1 edge scatter: 16 lanes per edge -> contiguous 64B atomic bursts,
//    xw/agg1/dinv are L2-resident (192MB L2 >> 13MB working set)
// ---------------------------------------------------------------------------
__global__ void k_scatter1(const int* __restrict__ src, const int* __restrict__ dst,
                           const float* __restrict__ dinv, const float* __restrict__ xw,
                           float* __restrict__ agg1) {
    unsigned t = blockIdx.x * blockDim.x + threadIdx.x;   // < E*16 = 51.2M
    int e = (int)(t >> 4);
    int k = (int)(t & 15);
    if (e >= NEDGES) return;
    int s = src[e], d = dst[e];
    float norm = dinv[s] * dinv[d];
    atomicAdd(&agg1[(size_t)d * HDIM + k], xw[(size_t)s * HDIM + k] * norm);
}

// ---------------------------------------------------------------------------
// 6) fused: self-loop add (non-atomic) + bias + ReLU + (h @ W2) -> hw [N,2]
// ---------------------------------------------------------------------------
__global__ void k_h_gemm2(const float* __restrict__ agg1, const float* __restrict__ xw,
                          const float* __restrict__ dinv, const float* __restrict__ b1,
                          const float* __restrict__ W2, float* __restrict__ hw) {
    int n = blockIdx.x * blockDim.x + threadIdx.x;
    if (n >= NNODES) return;
    float di = dinv[n];
    float sl = di * di;                       // self-loop norm = dinv*dinv
    float c0 = 0.0f, c1 = 0.0f;
#pragma unroll
    for (int k = 0; k < HDIM; ++k) {
        float h = agg1[(size_t)n * HDIM + k] + xw[(size_t)n * HDIM + k] * sl + b1[k];
        h = fmaxf(h, 0.0f);                   // ReLU (dropout is identity in eval)
        c0 = fmaf(h, W2[k * CDIM + 0], c0);
        c1 = fmaf(h, W2[k * CDIM + 1], c1);
    }
    hw[(size_t)n * CDIM + 0] = c0;
    hw[(size_t)n * CDIM + 1] = c1;
}

// ---------------------------------------------------------------------------
// 7) layer-2 edge scatter: 2 lanes per edge
// ---------------------------------------------------------------------------
__global__ void k_scatter2(const int* __restrict__ src, const int* __restrict__ dst,
                           const float* __restrict__ dinv, const float* __restrict__ hw,
                           float* __restrict__ agg2) {
    unsigned t = blockIdx.x * blockDim.x + threadIdx.x;   // < E*2 = 6.4M
    int e = (int)(t >> 1);
    int c = (int)(t & 1);
    if (e >= NEDGES) return;
    int s = src[e], d = dst[e];
    float norm = dinv[s] * dinv[d];
    atomicAdd(&agg2[(size_t)d * CDIM + c], hw[(size_t)s * CDIM + c] * norm);
}

// ---------------------------------------------------------------------------
// 8) finalize: self-loop + bias + log_softmax over C=2, write d_out
// ---------------------------------------------------------------------------
__global__ void k_final(const float* __restrict__ agg2, const float* __restrict__ hw,
                        const float* __restrict__ dinv, const float* __restrict__ b2,
                        float* __restrict__ out) {
    int n = blockIdx.x * blockDim.x + threadIdx.x;
    if (n >= NNODES) return;
    float di = dinv[n];
    float sl = di * di;
    float v0 = agg2[(size_t)n * CDIM + 0] + hw[(size_t)n * CDIM + 0] * sl + b2[0];
    float v1 = agg2[(size_t)n * CDIM + 1] + hw[(size_t)n * CDIM + 1] * sl + b2[1];
    float mx  = fmaxf(v0, v1);
    float lse = mx + logf(expf(v0 - mx) + expf(v1 - mx));
    out[(size_t)n * CDIM + 0] = v0 - lse;
    out[(size_t)n * CDIM + 1] = v1 - lse;
}

// ---------------------------------------------------------------------------
extern "C" void kernel_launch(void* const* d_in, const int* in_sizes, int n_in,
                              void* d_out, int out_size, void* d_ws, size_t ws_size,
                              hipStream_t stream) {
    const float* x  = (const float*)d_in[0];
    const int*   ei = (const int*)  d_in[1];
    const float* W1 = (const float*)d_in[2];
    const float* b1 = (const float*)d_in[3];
    const float* W2 = (const float*)d_in[4];
    const float* b2 = (const float*)d_in[5];
    const int* src = ei;            // edge_index[0]
    const int* dst = ei + NEDGES;   // edge_index[1]
    float* out = (float*)d_out;

    // workspace layout (floats): dinv[N] | xw[16N] | agg1[16N] | hw[2N] | agg2[2N]
    float* ws   = (float*)d_ws;
    float* dinv = ws;
    float* xw   = ws + (size_t)NNODES;
    float* agg1 = ws + (size_t)NNODES * 17;
    float* hw   = ws + (size_t)NNODES * 33;
    float* agg2 = ws + (size_t)NNODES * 35;   // total 37N floats ~= 14.8 MB

    const int B = 256;

    // 1) init (covers 16N elements; agg2/deg ranges are subsets)
    k_init<<<(NNODES * HDIM + B - 1) / B, B, 0, stream>>>(dinv, agg1, agg2);
    // 2) degree
    k_degree<<<(NEDGES + B - 1) / B, B, 0, stream>>>(dst, dinv);
    // 3) dinv = rsqrt(deg)
    k_rsqrt<<<(NNODES + B - 1) / B, B, 0, stream>>>(dinv);
    // 4) xw = x @ W1  (WMMA f32; 8 waves per block, one 16-row tile per wave)
    {
        int tiles = NNODES / 16;                    // 6250 (exact)
        int wavesPerBlock = B / 32;                 // 8
        int blocks = (tiles + wavesPerBlock - 1) / wavesPerBlock;
        k_gemm1<<<blocks, B, 0, stream>>>(x, W1, xw);
    }
    // 5) layer-1 scatter-add
    {
        unsigned total = (unsigned)NEDGES * HDIM;   // 51.2M
        k_scatter1<<<(total + B - 1) / B, B, 0, stream>>>(src, dst, dinv, xw, agg1);
    }
    // 6) self-loop + bias + relu + h@W2
    k_h_gemm2<<<(NNODES + B - 1) / B, B, 0, stream>>>(agg1, xw, dinv, b1, W2, hw);
    // 7) layer-2 scatter-add
    {
        unsigned total = (unsigned)NEDGES * CDIM;   // 6.4M
        k_scatter2<<<(total + B - 1) / B, B, 0, stream>>>(src, dst, dinv, hw, agg2);
    }
    // 8) finalize: self-loop + bias + log_softmax
    k_final<<<(NNODES + B - 1) / B, B, 0, stream>>>(agg2, hw, dinv, b2, out);
}